// ButterworthFilter_72825465471259
// MI455X (gfx1250) — compile-verified
//
#include <hip/hip_runtime.h>
#include <cmath>
#include <complex>

#ifndef M_PI
#define M_PI 3.14159265358979323846
#endif

#define T_LEN   65536
#define NCH     256
#define LC      256                 // chunk length (time samples per chunk)
#define NCHUNK  (T_LEN / LC)        // 256 chunks per channel

typedef __attribute__((ext_vector_type(2))) float v2f;
typedef __attribute__((ext_vector_type(4))) float v4f;
typedef __attribute__((ext_vector_type(8))) float v8f;

#if __has_builtin(__builtin_amdgcn_wmma_f32_16x16x4_f32)
#define HAVE_WMMA_F32X4 1
#else
#define HAVE_WMMA_F32X4 0
#endif

struct Coef10 { float b[5]; float a[5]; };
struct Mat16  { float m[16]; };                       // M^LC row-major
struct P3C    { float Af[4][32][2]; float Hf[32][2]; float b[5]; float a[5]; };
#define P3C_FLOATS ((int)(sizeof(P3C) / sizeof(float)))

// ---------------------------------------------------------------------------
// Pass 1: per-chunk particular end-state from zero initial state (scalar DF2T)
// grid: NCH blocks x NCHUNK threads; thread = (channel, chunk)
// ---------------------------------------------------------------------------
__global__ void iir_pass1_state(const float* __restrict__ x,
                                float* __restrict__ zend, Coef10 C) {
    const int ch = blockIdx.x;
    const int chunk = threadIdx.x;
    const float* xp = x + (size_t)ch * T_LEN + (size_t)chunk * LC;
    const float b0 = C.b[0], b1 = C.b[1], b2 = C.b[2], b3 = C.b[3], b4 = C.b[4];
    const float a1 = C.a[1], a2 = C.a[2], a3 = C.a[3], a4 = C.a[4];
    float z0 = 0.f, z1 = 0.f, z2 = 0.f, z3 = 0.f;
    for (int i = 0; i < LC; i += 4) {
        const float4 xv = *reinterpret_cast<const float4*>(xp + i);
        float xs[4] = {xv.x, xv.y, xv.z, xv.w};
#pragma unroll
        for (int j = 0; j < 4; ++j) {
            const float xx = xs[j];
            const float y = b0 * xx + z0;
            z0 = z1 + b1 * xx - a1 * y;
            z1 = z2 + b2 * xx - a2 * y;
            z2 = z3 + b3 * xx - a3 * y;
            z3 =      b4 * xx - a4 * y;
        }
    }
    float4 zo; zo.x = z0; zo.y = z1; zo.z = z2; zo.w = z3;
    *reinterpret_cast<float4*>(zend + ((size_t)ch * NCHUNK + chunk) * 4) = zo;
}

// ---------------------------------------------------------------------------
// Pass 2: serial scan over chunks: z_start[k] = M^LC * z_start[k-1] + zend0[k-1]
// grid: 1 block x NCH threads (thread = channel)
// ---------------------------------------------------------------------------
__global__ void iir_pass2_scan(const float* __restrict__ zend,
                               float* __restrict__ zstart, Mat16 M) {
    const int ch = threadIdx.x;
    if (ch >= NCH) return;
    float z0 = 0.f, z1 = 0.f, z2 = 0.f, z3 = 0.f;
    for (int k = 0; k < NCHUNK; ++k) {
        const size_t o = ((size_t)ch * NCHUNK + k) * 4;
        float4 zo; zo.x = z0; zo.y = z1; zo.z = z2; zo.w = z3;
        *reinterpret_cast<float4*>(zstart + o) = zo;
        const float4 e = *reinterpret_cast<const float4*>(zend + o);
        const float n0 = M.m[0]  * z0 + M.m[1]  * z1 + M.m[2]  * z2 + M.m[3]  * z3 + e.x;
        const float n1 = M.m[4]  * z0 + M.m[5]  * z1 + M.m[6]  * z2 + M.m[7]  * z3 + e.y;
        const float n2 = M.m[8]  * z0 + M.m[9]  * z1 + M.m[10] * z2 + M.m[11] * z3 + e.z;
        const float n3 = M.m[12] * z0 + M.m[13] * z1 + M.m[14] * z2 + M.m[15] * z3 + e.w;
        z0 = n0; z1 = n1; z2 = n2; z3 = n3;
    }
}

// ---------------------------------------------------------------------------
// Upload per-lane WMMA fragment constants into workspace
// ---------------------------------------------------------------------------
__global__ void iir_upload_consts(P3C C, float* __restrict__ dst) {
    const int i = threadIdx.x;
    const float* p = reinterpret_cast<const float*>(&C);
    if (i < P3C_FLOATS) dst[i] = p[i];
}

// ---------------------------------------------------------------------------
// Pass 3: WMMA tile filter. One wave per (16-channel group, chunk).
// Sub-block of 16 timesteps:  Y = L*X + H*Z0  via 5x V_WMMA_F32_16X16X4_F32.
// State carried across sub-blocks from last 4 x/y via DF2T window identity.
// grid: 512 blocks x 256 threads (8 waves/block, 4096 waves total)
// ---------------------------------------------------------------------------
__global__ void iir_pass3_wmma(const float* __restrict__ x,
                               const float* __restrict__ zstart,
                               const float* __restrict__ cws,
                               float* __restrict__ out) {
#if HAVE_WMMA_F32X4
    const int wid   = blockIdx.x * (blockDim.x >> 5) + (threadIdx.x >> 5);
    const int lane  = threadIdx.x & 31;
    const int chunk = wid & (NCHUNK - 1);
    const int grp   = wid >> 8;                 // NCHUNK == 256
    const int c     = lane & 15;
    const int half  = lane >> 4;                // 0: rows 0-7 / K 0,1 ; 1: rows 8-15 / K 2,3
    const int ch    = grp * 16 + c;

    // Per-lane fragment constants (A-layout: row = lane%16, K pair by half-wave)
    v2f Af0 = *reinterpret_cast<const v2f*>(cws + (0 * 32 + lane) * 2);
    v2f Af1 = *reinterpret_cast<const v2f*>(cws + (1 * 32 + lane) * 2);
    v2f Af2 = *reinterpret_cast<const v2f*>(cws + (2 * 32 + lane) * 2);
    v2f Af3 = *reinterpret_cast<const v2f*>(cws + (3 * 32 + lane) * 2);
    v2f Hf  = *reinterpret_cast<const v2f*>(cws + 256 + lane * 2);
    const float* ba = cws + 320;                // b[5] then a[5]
    const float b1 = ba[1], b2 = ba[2], b3 = ba[3], b4 = ba[4];
    const float a1 = ba[6], a2 = ba[7], a3 = ba[8], a4 = ba[9];

    // Initial state fragment (B-layout: K = 2*half + {0,1}, N = channel lane)
    v2f Zb = *reinterpret_cast<const v2f*>(
        zstart + ((size_t)ch * NCHUNK + chunk) * 4 + half * 2);

    const float* xp = x   + (size_t)ch * T_LEN + (size_t)chunk * LC + half * 2;
    float*       op = out + (size_t)ch * T_LEN + (size_t)chunk * LC + half * 8;

    for (int sb = 0; sb < LC / 16; ++sb) {
        const float* xq = xp + sb * 16;
        const v2f X0 = *reinterpret_cast<const v2f*>(xq + 0);
        const v2f X1 = *reinterpret_cast<const v2f*>(xq + 4);
        const v2f X2 = *reinterpret_cast<const v2f*>(xq + 8);
        const v2f X3 = *reinterpret_cast<const v2f*>(xq + 12);

        v8f acc = {};
        acc = __builtin_amdgcn_wmma_f32_16x16x4_f32(false, Hf,  false, Zb, (short)0, acc, false, false);
        acc = __builtin_amdgcn_wmma_f32_16x16x4_f32(false, Af0, false, X0, (short)0, acc, false, false);
        acc = __builtin_amdgcn_wmma_f32_16x16x4_f32(false, Af1, false, X1, (short)0, acc, false, false);
        acc = __builtin_amdgcn_wmma_f32_16x16x4_f32(false, Af2, false, X2, (short)0, acc, false, false);
        acc = __builtin_amdgcn_wmma_f32_16x16x4_f32(false, Af3, false, X3, (short)0, acc, false, false);

        // Store: lane holds y[t0 + r + 8*half], r = 0..7 (contiguous 32B)
        v4f lo = __builtin_shufflevector(acc, acc, 0, 1, 2, 3);
        v4f hi = __builtin_shufflevector(acc, acc, 4, 5, 6, 7);
        *reinterpret_cast<v4f*>(op + sb * 16 + 0) = lo;
        *reinterpret_cast<v4f*>(op + sb * 16 + 4) = hi;

        // State exiting the sub-block (valid on high half-wave lanes):
        // z[i] = sum_{j=0..3-i} ( b[i+1+j]*x[15-j] - a[i+1+j]*y[15-j] )
        const float x12 = __shfl(X3.x, c, 32);          // from low lane c
        const float x13 = __shfl(X3.y, c, 32);
        const float x14 = X3.x, x15 = X3.y;             // high lanes' own
        const float y12 = acc[4], y13 = acc[5], y14 = acc[6], y15 = acc[7];
        const float z3 = b4 * x15 - a4 * y15;
        const float z2 = b3 * x15 - a3 * y15 + b4 * x14 - a4 * y14;
        const float z1 = b2 * x15 - a2 * y15 + b3 * x14 - a3 * y14 + b4 * x13 - a4 * y13;
        const float z0 = b1 * x15 - a1 * y15 + b2 * x14 - a2 * y14
                       + b3 * x13 - a3 * y13 + b4 * x12 - a4 * y12;
        const float v0 = __shfl(z0, c + 16, 32);        // broadcast from high lane
        const float v1 = __shfl(z1, c + 16, 32);
        Zb.x = half ? z2 : v0;
        Zb.y = half ? z3 : v1;
    }
#else
    // Fallback: scalar recurrence per (channel, chunk) seeded by zstart.
    const int tid = blockIdx.x * blockDim.x + threadIdx.x;
    if (tid >= NCH * NCHUNK) return;
    const int ch = tid >> 8, chunk = tid & (NCHUNK - 1);
    const float* ba = cws + 320;
    const float b0 = ba[0], b1 = ba[1], b2 = ba[2], b3 = ba[3], b4 = ba[4];
    const float a1 = ba[6], a2 = ba[7], a3 = ba[8], a4 = ba[9];
    const float* zs = zstart + ((size_t)ch * NCHUNK + chunk) * 4;
    float z0 = zs[0], z1 = zs[1], z2 = zs[2], z3 = zs[3];
    const float* xp = x   + (size_t)ch * T_LEN + (size_t)chunk * LC;
    float*       op = out + (size_t)ch * T_LEN + (size_t)chunk * LC;
    for (int i = 0; i < LC; ++i) {
        const float xx = xp[i];
        const float y = b0 * xx + z0;
        z0 = z1 + b1 * xx - a1 * y;
        z1 = z2 + b2 * xx - a2 * y;
        z2 = z3 + b3 * xx - a3 * y;
        z3 =      b4 * xx - a4 * y;
        op[i] = y;
    }
#endif
}

// ---------------------------------------------------------------------------
// Host: coefficient design (identical math to reference) + fragment precompute
// ---------------------------------------------------------------------------
static void design_butter(float bf[5], float af[5]) {
    const int N = 4;
    const double WN = 0.5, fs2 = 4.0;
    const double warped = fs2 * std::tan(M_PI * WN / 4.0);
    std::complex<double> p[4], pd[4], prod(1.0, 0.0);
    for (int k = 1; k <= N; ++k)
        p[k - 1] = warped * std::exp(std::complex<double>(0.0, M_PI * (2 * k + N - 1) / (2.0 * N)));
    const double gain = warped * warped * warped * warped;
    for (int i = 0; i < N; ++i) { pd[i] = (fs2 + p[i]) / (fs2 - p[i]); prod *= (fs2 - p[i]); }
    const double kd = gain / prod.real();
    const double binom[5] = {1, 4, 6, 4, 1};
    for (int i = 0; i < 5; ++i) bf[i] = (float)(kd * binom[i]);
    std::complex<double> coef[5] = {1, 0, 0, 0, 0};
    for (int i = 0; i < N; ++i) {
        std::complex<double> nc[5] = {0, 0, 0, 0, 0};
        for (int j = 0; j <= i; ++j) { nc[j] += coef[j]; nc[j + 1] -= coef[j] * pd[i]; }
        for (int j = 0; j <= i + 1; ++j) coef[j] = nc[j];
    }
    for (int i = 0; i < 5; ++i) af[i] = (float)coef[i].real();
}

static void mat4_mul(const double A[4][4], const double B[4][4], double C[4][4]) {
    double T[4][4];
    for (int i = 0; i < 4; ++i)
        for (int j = 0; j < 4; ++j) {
            double s = 0;
            for (int k = 0; k < 4; ++k) s += A[i][k] * B[k][j];
            T[i][j] = s;
        }
    for (int i = 0; i < 4; ++i) for (int j = 0; j < 4; ++j) C[i][j] = T[i][j];
}

extern "C" void kernel_launch(void* const* d_in, const int* in_sizes, int n_in,
                              void* d_out, int out_size, void* d_ws, size_t ws_size,
                              hipStream_t stream) {
    (void)in_sizes; (void)n_in; (void)out_size; (void)ws_size;
    const float* x = (const float*)d_in[0];
    float* out = (float*)d_out;
    float* ws  = (float*)d_ws;
    float* zend   = ws;                               // NCH*NCHUNK*4 floats (1 MB)
    float* zstart = ws + (size_t)NCH * NCHUNK * 4;    // 1 MB
    float* cws    = zstart + (size_t)NCH * NCHUNK * 4;// fragment constants

    // --- filter design (f32-rounded, matching reference) ---
    float bf[5], af[5];
    design_butter(bf, af);

    // --- state-space: z' = M z + v x ; y = e0.z + b0 x  (built from f32 coefs) ---
    double Md[4][4], vd[4];
    for (int i = 0; i < 4; ++i)
        for (int j = 0; j < 4; ++j)
            Md[i][j] = ((j == i + 1) ? 1.0 : 0.0) - (double)af[i + 1] * (j == 0 ? 1.0 : 0.0);
    for (int i = 0; i < 4; ++i) vd[i] = (double)bf[i + 1] - (double)af[i + 1] * (double)bf[0];

    // impulse response h[d] = (M^d v)[0], rows Hrow[t] = e0^T M^t
    double h[16], u[4] = {vd[0], vd[1], vd[2], vd[3]};
    for (int d = 0; d < 16; ++d) {
        h[d] = u[0];
        double nu[4];
        for (int i = 0; i < 4; ++i) {
            double s = 0; for (int k = 0; k < 4; ++k) s += Md[i][k] * u[k];
            nu[i] = s;
        }
        for (int i = 0; i < 4; ++i) u[i] = nu[i];
    }
    double Hrow[16][4], r[4] = {1, 0, 0, 0};
    for (int t = 0; t < 16; ++t) {
        for (int k = 0; k < 4; ++k) Hrow[t][k] = r[k];
        double nr[4];
        for (int j = 0; j < 4; ++j) {
            double s = 0; for (int k = 0; k < 4; ++k) s += r[k] * Md[k][j];
            nr[j] = s;
        }
        for (int j = 0; j < 4; ++j) r[j] = nr[j];
    }
    double Lmat[16][16];
    for (int t = 0; t < 16; ++t)
        for (int s = 0; s < 16; ++s)
            Lmat[t][s] = (t == s) ? (double)bf[0] : (t > s ? h[t - 1 - s] : 0.0);

    // M^LC via 8 squarings (LC = 256)
    double P[4][4];
    for (int i = 0; i < 4; ++i) for (int j = 0; j < 4; ++j) P[i][j] = Md[i][j];
    for (int s = 0; s < 8; ++s) mat4_mul(P, P, P);

    // --- pack kernel-arg structs ---
    Coef10 c10;
    for (int i = 0; i < 5; ++i) { c10.b[i] = bf[i]; c10.a[i] = af[i]; }
    Mat16 m16;
    for (int i = 0; i < 4; ++i) for (int j = 0; j < 4; ++j) m16.m[i * 4 + j] = (float)P[i][j];

    P3C p3;
    for (int q = 0; q < 4; ++q)
        for (int lane = 0; lane < 32; ++lane) {
            const int row = lane & 15, kb = 4 * q + (lane >> 4) * 2;
            p3.Af[q][lane][0] = (float)Lmat[row][kb];
            p3.Af[q][lane][1] = (float)Lmat[row][kb + 1];
        }
    for (int lane = 0; lane < 32; ++lane) {
        const int row = lane & 15, kb = (lane >> 4) * 2;
        p3.Hf[lane][0] = (float)Hrow[row][kb];
        p3.Hf[lane][1] = (float)Hrow[row][kb + 1];
    }
    for (int i = 0; i < 5; ++i) { p3.b[i] = bf[i]; p3.a[i] = af[i]; }

    // --- launch pipeline (all on stream; stream order gives dependencies) ---
    iir_upload_consts<<<1, 512, 0, stream>>>(p3, cws);
    iir_pass1_state<<<NCH, NCHUNK, 0, stream>>>(x, zend, c10);
    iir_pass2_scan<<<1, NCH, 0, stream>>>(zend, zstart, m16);
    iir_pass3_wmma<<<(NCH / 16) * NCHUNK / 8, 256, 0, stream>>>(x, zstart, cws, out);
}